// GIN_3023656976826
// MI455X (gfx1250) — compile-verified
//
#include <hip/hip_runtime.h>
#include <hip/hip_bf16.h>

// ---------------------------------------------------------------------------
// GIN (3-layer) for MI455X / gfx1250, wave32.
//   agg = h  (copy, folds the "+x" self term)
//   agg[dst] += h[src]  (atomic f32 scatter, one wave per edge)
//   h_next = relu(agg @ W + b)  (WMMA f32 16x16x4 GEMM, LDS-staged A tile)
// Round 2: N is a compile-time template param so all W loads / C stores use
// constant immediate offsets (no 64-bit muls in the WMMA inner loop); all
// index math is 32-bit.
// ---------------------------------------------------------------------------

#define N_NODES 100000
#define N_EDGES 600000
#define DIM     128

typedef float v2f __attribute__((ext_vector_type(2)));
typedef float v8f __attribute__((ext_vector_type(8)));

// ---------------------------------------------------------------- copy ------
__global__ void gin_copy_f4(const float4* __restrict__ src,
                            float4* __restrict__ dst, int n4) {
    int i = blockIdx.x * blockDim.x + threadIdx.x;
    int stride = gridDim.x * blockDim.x;
    for (; i < n4; i += stride) dst[i] = src[i];
}

// --------------------------------------------------------- scatter-add -----
// One wave (32 lanes) per edge: lanes cover 128 features as float4 each.
// Gather h[src] is one coalesced 512B line per wave; accumulate with
// global_atomic_add_f32 into agg[dst]. 32-bit offsets (51MB buffers).
__global__ void gin_scatter_add(const float* __restrict__ h,
                                const int* __restrict__ src,
                                const int* __restrict__ dst,
                                float* __restrict__ agg, int nEdges) {
    int gid  = blockIdx.x * blockDim.x + threadIdx.x;
    int edge = gid >> 5;
    int lane = threadIdx.x & 31;
    if (edge >= nEdges) return;
    unsigned s = (unsigned)src[edge];
    unsigned d = (unsigned)dst[edge];
    float4 v = *(const float4*)(h + (s << 7) + lane * 4);
    float* out = agg + (d << 7) + lane * 4;
    atomicAdd(out + 0, v.x);
    atomicAdd(out + 1, v.y);
    atomicAdd(out + 2, v.z);
    atomicAdd(out + 3, v.w);
}

// ---------------------------------------------------------------- GEMM -----
// C[M x N] = A[M x 128] * W[128 x N] + bias, optional ReLU.
// Grid: M/16 blocks. Block: (N/16) waves * 32 threads; wave w owns the
// 16x16 output tile at (blockIdx.x*16, w*16).
//
// V_WMMA_F32_16X16X4_F32 fragment layouts (ISA 7.12.2, wave32):
//   A (16x4):  v0 = A[m][k+2*half], v1 = A[m][k+2*half+1]; m = lane&15,
//              half = lane>>4.
//   B (4x16):  v0 = W[k+2*half][n], v1 = W[k+2*half+1][n]; n = col0+(lane&15).
//   C/D:       vgpr r -> row (r + 8*half), col = col0+(lane&15).
template <int N, bool RELU>
__global__ void gin_gemm_wmma(const float* __restrict__ A,
                              const float* __restrict__ W,
                              const float* __restrict__ bias,
                              float* __restrict__ C) {
    __shared__ float As[16 * DIM];   // 8 KB A tile shared by all waves

    const int tid  = threadIdx.x;
    const int lane = tid & 31;
    const int wave = tid >> 5;
    const unsigned row0 = blockIdx.x * 16;

    // cooperative stage of the 16x128 A tile into LDS (float4 stores)
    for (int i = tid * 4; i < 16 * DIM; i += blockDim.x * 4) {
        *(float4*)(&As[i]) = *(const float4*)(A + (row0 << 7) + i);
    }
    __syncthreads();

    const int half = lane >> 4;      // 0: lanes 0-15, 1: lanes 16-31
    const int lrow = lane & 15;
    const int col0 = wave * 16;

    // Per-lane base pointers; inside the unrolled loop every access is a
    // compile-time constant byte offset from these.
    const float* __restrict__ Wp = W + (half * 2) * N + col0 + lrow;   // row k+2*half, col n
    const float* __restrict__ Ap = As + lrow * DIM + half * 2;

    v8f acc = {};
#pragma unroll
    for (int k = 0; k < DIM; k += 4) {
        v2f a = *(const v2f*)(Ap + k);           // ds_load_b64, imm offset
        v2f b;
        b.x = Wp[k * N];                         // global_load_b32, imm offset
        b.y = Wp[k * N + N];
        acc = __builtin_amdgcn_wmma_f32_16x16x4_f32(
            /*neg_a=*/false, a, /*neg_b=*/false, b,
            /*c_mod=*/(short)0, acc, /*reuse_a=*/false, /*reuse_b=*/false);
    }

    const float bv = bias[col0 + lrow];
    float* __restrict__ Cp = C + (row0 + half * 8) * N + col0 + lrow;
#pragma unroll
    for (int r = 0; r < 8; ++r) {
        float v = acc[r] + bv;
        if (RELU) v = v > 0.0f ? v : 0.0f;
        Cp[r * N] = v;                           // global_store_b32, imm offset
    }
}

// ------------------------------------------------------------- launch ------
extern "C" void kernel_launch(void* const* d_in, const int* in_sizes, int n_in,
                              void* d_out, int out_size, void* d_ws, size_t ws_size,
                              hipStream_t stream) {
    const float* x   = (const float*)d_in[0];
    const int*   ei  = (const int*)d_in[1];     // [2, E]
    const float* W1  = (const float*)d_in[2];
    const float* b1  = (const float*)d_in[3];
    const float* W2  = (const float*)d_in[4];
    const float* b2  = (const float*)d_in[5];
    const float* W3  = (const float*)d_in[6];
    const float* b3  = (const float*)d_in[7];
    float*       out = (float*)d_out;

    const int* src = ei;
    const int* dst = ei + N_EDGES;

    const size_t featElems = (size_t)N_NODES * DIM;      // 12.8M floats
    float* bufA = (float*)d_ws;
    float* bufB = bufA + featElems;
    float* bufC = bufB + featElems;
    (void)ws_size; (void)n_in; (void)in_sizes; (void)out_size;

    const int n4 = (int)(featElems / 4);
    const int copyBlocks = (n4 + 255) / 256;
    const int scatBlocks = (N_EDGES * 32 + 255) / 256;   // 1 wave per edge
    const int gemmBlocks = N_NODES / 16;                 // 6250

    // ---- layer 1: agg1 = x + scatter(x); h1 = relu(agg1 @ W1 + b1) -> bufB
    gin_copy_f4<<<copyBlocks, 256, 0, stream>>>((const float4*)x, (float4*)bufA, n4);
    gin_scatter_add<<<scatBlocks, 256, 0, stream>>>(x, src, dst, bufA, N_EDGES);
    gin_gemm_wmma<DIM, true><<<gemmBlocks, 256, 0, stream>>>(bufA, W1, b1, bufB);

    // ---- layer 2: agg2 = h1 + scatter(h1); h2 = relu(agg2 @ W2 + b2) -> bufA
    gin_copy_f4<<<copyBlocks, 256, 0, stream>>>((const float4*)bufB, (float4*)bufC, n4);
    gin_scatter_add<<<scatBlocks, 256, 0, stream>>>(bufB, src, dst, bufC, N_EDGES);
    gin_gemm_wmma<DIM, true><<<gemmBlocks, 256, 0, stream>>>(bufC, W2, b2, bufA);

    // ---- layer 3: agg3 = h2 + scatter(h2); out = agg3 @ W3 + b3  (N=64)
    gin_copy_f4<<<copyBlocks, 256, 0, stream>>>((const float4*)bufA, (float4*)bufB, n4);
    gin_scatter_add<<<scatBlocks, 256, 0, stream>>>(bufA, src, dst, bufB, N_EDGES);
    gin_gemm_wmma<64, false><<<gemmBlocks, 128, 0, stream>>>(bufB, W3, b3, out);
}